// OctreeAttention_29274497089726
// MI455X (gfx1250) — compile-verified
//
#include <hip/hip_runtime.h>
#include <hip/hip_bf16.h>
#include <math.h>

// ---------------- problem constants ----------------
#define CDIM   256
#define HEADS  8
#define HD     32
#define KPTS   32
#define GCTX   8
#define WTOK   40            // KPTS + GCTX
#define WPAD   48            // 40 padded to 3x16
#define NW     4096
#define TROWS  (NW * WTOK)   // 163840
#define POS_BND 25
#define RPE_NUM 51
#define QSCALE  0.17677669529663687f   // 32^-0.5

typedef __attribute__((ext_vector_type(2))) float v2f;
typedef __attribute__((ext_vector_type(8))) float v8f;
typedef __attribute__((ext_vector_type(4))) int   v4i;

__device__ __forceinline__ v8f wmma_f32_4(v2f a, v2f b, v8f c) {
    // D = A(16x4) * B(4x16) + C(16x16), fp32 WMMA
    return __builtin_amdgcn_wmma_f32_16x16x4_f32(false, a, false, b, (short)0, c, false, false);
}

// -------- CDNA5 async global->LDS copy (ASYNCcnt-tracked), guarded --------
#if defined(__has_builtin)
# if __has_builtin(__builtin_amdgcn_global_load_async_to_lds_b128)
#  define HAVE_ASYNC_COPY 1
# endif
#endif
#ifndef HAVE_ASYNC_COPY
# define HAVE_ASYNC_COPY 0
#endif

#if HAVE_ASYNC_COPY
typedef v4i __attribute__((address_space(1)))* glb_v4p;   // global int4*
typedef v4i __attribute__((address_space(3)))* lds_v4p;   // LDS int4*

__device__ __forceinline__ void async_cp16(const float* g, float* l) {
    __builtin_amdgcn_global_load_async_to_lds_b128(
        (glb_v4p)(uintptr_t)g, (lds_v4p)(uint32_t)(uintptr_t)l, 0, 0);
}
__device__ __forceinline__ void async_wait0() {
# if __has_builtin(__builtin_amdgcn_s_wait_asynccnt)
    __builtin_amdgcn_s_wait_asynccnt(0);
# else
    asm volatile("s_wait_asynccnt 0x0" ::: "memory");
# endif
}
#endif

// =====================================================================
// GEMM: C[M,N] = A[M,Kd] * Bw[Kd,N] + bias[N]; optional column sum/sumsq
// Block tile 64x128, BK=32, 256 threads = 8 waves (4 in M, 2 in N).
// Double-buffered LDS with async global->LDS copies when available.
// =====================================================================
template <bool STD>
__global__ __launch_bounds__(256) void gemm_f32_wmma(
    const float* __restrict__ A, const float* __restrict__ Bw,
    const float* __restrict__ bias, float* __restrict__ C,
    int M, int N, int Kd, float* __restrict__ gsum, float* __restrict__ gsqs)
{
    constexpr int BM = 64, BN = 128, BK = 32, APAD = 36;
    __shared__ float As[2][BM * APAD];   // padded: row stride 36 words
    __shared__ float Bs[2][BK * BN];
    __shared__ float sSum[BN];
    __shared__ float sSqs[BN];

    const int tid  = threadIdx.x;
    const int lane = tid & 31;
    const int wid  = tid >> 5;
    const int wm   = (wid & 3) * 16;     // wave row offset (16 rows)
    const int wn   = (wid >> 2) * 64;    // wave col offset (4 tiles of 16)
    const int bm   = blockIdx.y * BM;
    const int bn   = blockIdx.x * BN;

    const int lm = lane & 15;            // row/col within 16
    const int kh = (lane >> 4) * 2;      // K sub-offset 0 or 2

    if (STD && tid < BN) { sSum[tid] = 0.0f; sSqs[tid] = 0.0f; }

    auto issue_tile = [&](int buf, int k0) {
        #pragma unroll
        for (int i = tid; i < (BM * BK) / 4; i += 256) {   // A: 64x32
            int m  = i >> 3;
            int kq = (i & 7) << 2;
            const float* g = A + (size_t)(bm + m) * Kd + k0 + kq;
            float* l = &As[buf][m * APAD + kq];
#if HAVE_ASYNC_COPY
            async_cp16(g, l);
#else
            *(float4*)l = *(const float4*)g;
#endif
        }
        #pragma unroll
        for (int i = tid; i < (BK * BN) / 4; i += 256) {   // B: 32x128
            int kr = i >> 5;
            int nq = (i & 31) << 2;
            const float* g = Bw + (size_t)(k0 + kr) * N + bn + nq;
            float* l = &Bs[buf][kr * BN + nq];
#if HAVE_ASYNC_COPY
            async_cp16(g, l);
#else
            *(float4*)l = *(const float4*)g;
#endif
        }
    };

    v8f acc[4];
    #pragma unroll
    for (int j = 0; j < 4; ++j) acc[j] = (v8f)0.0f;

    const int nk = Kd / BK;
    issue_tile(0, 0);

    for (int kt = 0; kt < nk; ++kt) {
        const int buf = kt & 1;
#if HAVE_ASYNC_COPY
        async_wait0();
#endif
        __syncthreads();          // tile kt resident; prev compute finished everywhere
        if (kt + 1 < nk) issue_tile(buf ^ 1, (kt + 1) * BK);

        const float* Ab = &As[buf][0];
        const float* Bb = &Bs[buf][0];
        #pragma unroll
        for (int kk = 0; kk < BK; kk += 4) {
            v2f a = *(const v2f*)&Ab[(wm + lm) * APAD + kk + kh];
            #pragma unroll
            for (int j = 0; j < 4; ++j) {
                v2f b;
                b.x = Bb[(kk + kh)     * BN + wn + j * 16 + lm];
                b.y = Bb[(kk + kh + 1) * BN + wn + j * 16 + lm];
                acc[j] = wmma_f32_4(a, b, acc[j]);
            }
        }
    }

    // epilogue: bias add, store, optional column sum/sumsq
    const int rbase = bm + wm + ((lane < 16) ? 0 : 8);
    #pragma unroll
    for (int j = 0; j < 4; ++j) {
        const int cn = bn + wn + j * 16 + lm;
        const float bv = bias[cn];
        float ls = 0.0f, lq = 0.0f;
        #pragma unroll
        for (int r = 0; r < 8; ++r) {
            float v = acc[j][r] + bv;
            C[(size_t)(rbase + r) * N + cn] = v;
            if (STD) { ls += v; lq += v * v; }
        }
        if (STD) {
            atomicAdd(&sSum[wn + j * 16 + lm], ls);
            atomicAdd(&sSqs[wn + j * 16 + lm], lq);
        }
    }
    if (STD) {
        __syncthreads();
        if (tid < BN) {
            atomicAdd(&gsum[bn + tid], sSum[tid]);
            atomicAdd(&gsqs[bn + tid], sSqs[tid]);
        }
    }
}

// =====================================================================
// Fused windowed attention: 1 block = 1 window, wave h = head h.
// S = Q*Kt -> (scale + RPE + mask, softmax in LDS) -> O = P*V
// Pad rows/cols (40..47) are never consumed, so boundary handling is a
// branch-free row-index clamp (no exec-mask divergence around WMMA).
// =====================================================================
__global__ __launch_bounds__(256) void attn_wmma(
    const float* __restrict__ qkv, const int* __restrict__ rel_pos,
    const float* __restrict__ mask, const float* __restrict__ rpe_table,
    float* __restrict__ ctx)
{
    __shared__ float sS[HEADS * WPAD * WPAD];   // 8 * 48*48 f32 = 72KB

    const int n    = blockIdx.x;
    const int lane = threadIdx.x & 31;
    const int h    = threadIdx.x >> 5;
    float* S = &sS[h * WPAD * WPAD];

    const float* base = qkv + (size_t)n * WTOK * (3 * CDIM);
    const int qoff = h * HD;                 // q at +0
    const int koff = CDIM + h * HD;          // k at +256
    const int voff = 2 * CDIM + h * HD;      // v at +512

    const int lm = lane & 15;
    const int kh = (lane >> 4) * 2;          // 0 or 2

    // clamped row indices (pad rows read row 39; results unused)
    int rQ[3], rK[3];
    #pragma unroll
    for (int t = 0; t < 3; ++t) {
        int m = t * 16 + lm;
        int mc = (m < WTOK) ? m : (WTOK - 1);
        rQ[t] = mc * (3 * CDIM) + qoff;
        rK[t] = mc * (3 * CDIM) + koff;
    }

    // ---------------- Phase 1: S = Q * K^T  (40x40 in 3x3 tiles)
    v8f accS[9];
    #pragma unroll
    for (int t = 0; t < 9; ++t) accS[t] = (v8f)0.0f;

    for (int k0 = 0; k0 < HD; k0 += 4) {
        v2f aF[3], bF[3];
        #pragma unroll
        for (int tm = 0; tm < 3; ++tm)
            aF[tm] = *(const v2f*)&base[rQ[tm] + k0 + kh];
        #pragma unroll
        for (int tn = 0; tn < 3; ++tn)
            bF[tn] = *(const v2f*)&base[rK[tn] + k0 + kh];
        #pragma unroll
        for (int tm = 0; tm < 3; ++tm)
            #pragma unroll
            for (int tn = 0; tn < 3; ++tn)
                accS[tm * 3 + tn] = wmma_f32_4(aF[tm], bF[tn], accS[tm * 3 + tn]);
    }
    // dump S to LDS
    #pragma unroll
    for (int tm = 0; tm < 3; ++tm)
        #pragma unroll
        for (int tn = 0; tn < 3; ++tn)
            #pragma unroll
            for (int r = 0; r < 8; ++r) {
                int m = tm * 16 + r + ((lane < 16) ? 0 : 8);
                S[m * WPAD + tn * 16 + lm] = accS[tm * 3 + tn][r];
            }

    // ---------------- Phase 2: scale + RPE + mask, row softmax (in LDS)
    const float* mrow = mask + (size_t)n * WTOK * WTOK;
    const int*   rp   = rel_pos + (size_t)n * KPTS * KPTS * 3;
    for (int m = lane; m < WTOK; m += 32) {
        float mx = -1e30f;
        for (int j = 0; j < WTOK; ++j) {
            float val = S[m * WPAD + j] * QSCALE + mrow[m * WTOK + j];
            if (m >= GCTX && j >= GCTX) {
                const int* rr = rp + (((m - GCTX) * KPTS) + (j - GCTX)) * 3;
                #pragma unroll
                for (int a = 0; a < 3; ++a) {
                    int o = rr[a];
                    o = (o < -POS_BND) ? -POS_BND : ((o > POS_BND) ? POS_BND : o);
                    val += rpe_table[(o + POS_BND + a * RPE_NUM) * HEADS + h];
                }
            }
            S[m * WPAD + j] = val;
            mx = fmaxf(mx, val);
        }
        float sum = 0.0f;
        for (int j = 0; j < WTOK; ++j) {
            float p = __expf(S[m * WPAD + j] - mx);
            S[m * WPAD + j] = p;
            sum += p;
        }
        float inv = 1.0f / sum;
        for (int j = 0; j < WTOK; ++j) S[m * WPAD + j] *= inv;
    }
    // lanes reconverge; S[h] is private to this wave; wave-level LDS is in-order

    // ---------------- Phase 3: O = P * V   (40x40 * 40x32, K'=40 exact)
    v8f accO[6];
    #pragma unroll
    for (int t = 0; t < 6; ++t) accO[t] = (v8f)0.0f;

    for (int k0 = 0; k0 < WTOK; k0 += 4) {
        v2f aF[3], bF[2];
        #pragma unroll
        for (int tm = 0; tm < 3; ++tm)
            aF[tm] = *(const v2f*)&S[(tm * 16 + lm) * WPAD + k0 + kh];
        #pragma unroll
        for (int tn = 0; tn < 2; ++tn) {
            int d  = tn * 16 + lm;
            int wv = k0 + kh;
            bF[tn].x = base[wv       * (3 * CDIM) + voff + d];
            bF[tn].y = base[(wv + 1) * (3 * CDIM) + voff + d];
        }
        #pragma unroll
        for (int tm = 0; tm < 3; ++tm)
            #pragma unroll
            for (int tn = 0; tn < 2; ++tn)
                accO[tm * 2 + tn] = wmma_f32_4(aF[tm], bF[tn], accO[tm * 2 + tn]);
    }

    // store O -> ctx[(n*40 + w)*256 + h*32 + d]
    float* cbase = ctx + (size_t)n * WTOK * CDIM + h * HD;
    #pragma unroll
    for (int tm = 0; tm < 3; ++tm)
        #pragma unroll
        for (int tn = 0; tn < 2; ++tn)
            #pragma unroll
            for (int r = 0; r < 8; ++r) {
                int m = tm * 16 + r + ((lane < 16) ? 0 : 8);
                if (m < WTOK)
                    cbase[m * CDIM + tn * 16 + lm] = accO[tm * 2 + tn][r];
            }
}

// =====================================================================
// std finalize: unbiased (ddof=1)
// =====================================================================
__global__ void finalize_std(const float* __restrict__ gsum,
                             const float* __restrict__ gsqs,
                             float* __restrict__ out)
{
    int i = blockIdx.x * blockDim.x + threadIdx.x;
    if (i < 3 * CDIM) {
        const float nF = (float)TROWS;
        float mu  = gsum[i] / nF;
        float var = (gsqs[i] - nF * mu * mu) / (nF - 1.0f);
        out[i] = sqrtf(fmaxf(var, 0.0f));
    }
}

// =====================================================================
extern "C" void kernel_launch(void* const* d_in, const int* in_sizes, int n_in,
                              void* d_out, int out_size, void* d_ws, size_t ws_size,
                              hipStream_t stream)
{
    const float* data    = (const float*)d_in[0];
    const int*   rel_pos = (const int*)d_in[1];
    const float* mask    = (const float*)d_in[2];
    const float* qkv_w   = (const float*)d_in[3];
    const float* qkv_b   = (const float*)d_in[4];
    const float* proj_w  = (const float*)d_in[5];
    const float* proj_b  = (const float*)d_in[6];
    const float* rpe_tab = (const float*)d_in[7];
    float* out = (float*)d_out;

    // workspace layout
    float* qkv  = (float*)d_ws;                         // T x 768
    float* ctx  = qkv + (size_t)TROWS * (3 * CDIM);     // T x 256
    float* gsum = ctx + (size_t)TROWS * CDIM;           // 768
    float* gsqs = gsum + 3 * CDIM;                      // 768

    (void)hipMemsetAsync(gsum, 0, 2 * 3 * CDIM * sizeof(float), stream);

    // qkv = data @ qkv_w + qkv_b  (+ column sum/sumsq for std)
    gemm_f32_wmma<true><<<dim3((3 * CDIM) / 128, TROWS / 64), 256, 0, stream>>>(
        data, qkv_w, qkv_b, qkv, TROWS, 3 * CDIM, CDIM, gsum, gsqs);

    // fused windowed attention
    attn_wmma<<<NW, 256, 0, stream>>>(qkv, rel_pos, mask, rpe_tab, ctx);

    // out = ctx @ proj_w + proj_b
    gemm_f32_wmma<false><<<dim3(CDIM / 128, TROWS / 64), 256, 0, stream>>>(
        ctx, proj_w, proj_b, out, TROWS, CDIM, CDIM, nullptr, nullptr);

    // qkv_std appended after the (T, C) output
    finalize_std<<<3, 256, 0, stream>>>(gsum, gsqs, out + (size_t)TROWS * CDIM);
}